// RBF_21500606284259
// MI455X (gfx1250) — compile-verified
//
#include <hip/hip_runtime.h>

typedef float v2f __attribute__((ext_vector_type(2)));
typedef float v8f __attribute__((ext_vector_type(8)));

// B=65536, K=128, D=2. One wave = 32 points; block = 256 thr = 8 waves = 256 points.
__global__ __launch_bounds__(256) void RBF_kernel(
    const float* __restrict__ xin,        // (B,2)
    const float* __restrict__ centres,    // (K,2)
    const float* __restrict__ log_sigmas, // (K,)
    const float* __restrict__ alphas_uvx, // (K,1)
    const float* __restrict__ alphas_uvy, // (K,1)
    float* __restrict__ out)              // (B,2)
{
    __shared__ float2 s_cxy[128];
    __shared__ float4 s_meta[128];  // (|c|^2, s=exp(-2*ls), ax^2, ay)

    const int tid = threadIdx.x;
    if (tid < 128) {
        float cx = centres[2 * tid], cy = centres[2 * tid + 1];
        float s  = __expf(-2.0f * log_sigmas[tid]);
        float a  = alphas_uvx[tid];
        s_cxy[tid]  = make_float2(cx, cy);
        s_meta[tid] = make_float4(cx * cx + cy * cy, s, a * a, alphas_uvy[tid]);
    }
    __syncthreads();

    const int lane     = tid & 31;
    const int waveBase = blockIdx.x * 256 + (tid & ~31); // first point of this wave
    const bool isLo    = lane < 16;
    const int  h8      = (lane >> 1) & 8;                // 0 (lanes 0-15) or 8 (lanes 16-31)
    const int  hh      = (lane >> 4) & 1;                // half-wave id

    const float2 p2 = ((const float2*)xin)[waveBase + lane];
    const float px = p2.x, py = p2.y;

    // Per-lane view of the coordinates of the points this lane's C-slots map to.
    float pxA[8], pyA[8], pnA[8], pxB[8], pyB[8], pnB[8];
#pragma unroll
    for (int v = 0; v < 8; ++v) {
        pxA[v] = __shfl(px, h8 + v, 32);            // tile0 point m = v + 8*hh
        pyA[v] = __shfl(py, h8 + v, 32);
        pxB[v] = __shfl(px, 16 + h8 + v, 32);       // tile1 point m = 16 + v + 8*hh
        pyB[v] = __shfl(py, 16 + h8 + v, 32);
        pnA[v] = fmaf(pxA[v], pxA[v], pyA[v] * pyA[v]);
        pnB[v] = fmaf(pxB[v], pxB[v], pyB[v] * pyB[v]);
    }

    // A-matrix 16x4 f32 layout: VGPR0 = K0 (lanes 0-15) / K2 pad (16-31); VGPR1 = K1/K3 pad.
    v2f a0, a1;
    a0.x = isLo ? px : 0.0f;
    a0.y = isLo ? py : 0.0f;
    const float t1x = __shfl(px, (lane & 15) + 16, 32);
    const float t1y = __shfl(py, (lane & 15) + 16, 32);
    a1.x = isLo ? t1x : 0.0f;
    a1.y = isLo ? t1y : 0.0f;

    float S0[8], U0[8], V0[8], S1[8], U1[8], V1[8];
#pragma unroll
    for (int v = 0; v < 8; ++v) {
        S0[v] = 0.0f; U0[v] = 0.0f; V0[v] = 0.0f;
        S1[v] = 0.0f; U1[v] = 0.0f; V1[v] = 0.0f;
    }

    for (int chunk = 0; chunk < 8; ++chunk) {
        const int n = chunk * 16 + (lane & 15);
        const float2 c2 = s_cxy[n];
        const float4 m4 = s_meta[n];

        // B-matrix 4x16 f32: VGPR0 = K0 row (lanes 0-15) / K2 pad; VGPR1 = K1/K3 pad.
        v2f b;
        b.x = isLo ? c2.x : 0.0f;
        b.y = isLo ? c2.y : 0.0f;

        v8f cz = {};
        // dot[m][n] = px[m]*cx[n] + py[m]*cy[n], full f32 precision
        v8f d0 = __builtin_amdgcn_wmma_f32_16x16x4_f32(false, a0, false, b,
                                                       (short)0, cz, false, false);
        v8f d1 = __builtin_amdgcn_wmma_f32_16x16x4_f32(false, a1, false, b,
                                                       (short)0, cz, false, false);

        const float cn = m4.x, s = m4.y, ax2 = m4.z, aym = m4.w;
        const float s4 = 4.0f * s;

#pragma unroll
        for (int v = 0; v < 8; ++v) {
            { // tile 0 : point (v + 8*hh), centre n
                const float d2  = pnA[v] + cn - 2.0f * d0[v];   // consumes WMMA result
                const float phi = __expf(-s * d2);
                const float q   = ax2 * phi * phi;
                const float y   = __expf(q);
                const float c4  = s4 * q * y;
                const float t   = s4 * (1.0f + q);
                const float dx  = pxA[v] - c2.x, dy = pyA[v] - c2.y;
                const float uu  = c4 * fmaf(t, dy * (dx - dy), 1.0f);
                const float vv  = c4 * fmaf(t, dx * (dy - dx), 1.0f);
                const float e   = __expf(phi * aym);
                S0[v] += e;
                U0[v]  = fmaf(uu, e, U0[v]);
                V0[v]  = fmaf(vv, e, V0[v]);
            }
            { // tile 1 : point (16 + v + 8*hh), centre n
                const float d2  = pnB[v] + cn - 2.0f * d1[v];
                const float phi = __expf(-s * d2);
                const float q   = ax2 * phi * phi;
                const float y   = __expf(q);
                const float c4  = s4 * q * y;
                const float t   = s4 * (1.0f + q);
                const float dx  = pxB[v] - c2.x, dy = pyB[v] - c2.y;
                const float uu  = c4 * fmaf(t, dy * (dx - dy), 1.0f);
                const float vv  = c4 * fmaf(t, dx * (dy - dx), 1.0f);
                const float e   = __expf(phi * aym);
                S1[v] += e;
                U1[v]  = fmaf(uu, e, U1[v]);
                V1[v]  = fmaf(vv, e, V1[v]);
            }
        }
    }

    // Centre-dimension reduction: sum over the 16 lanes of each half-wave.
#pragma unroll
    for (int v = 0; v < 8; ++v) {
#pragma unroll
        for (int m = 1; m <= 8; m <<= 1) {
            S0[v] += __shfl_xor(S0[v], m, 32);
            U0[v] += __shfl_xor(U0[v], m, 32);
            V0[v] += __shfl_xor(V0[v], m, 32);
            S1[v] += __shfl_xor(S1[v], m, 32);
            U1[v] += __shfl_xor(U1[v], m, 32);
            V1[v] += __shfl_xor(V1[v], m, 32);
        }
    }

    // Writers: lane v (hh=0) and lane v+16 (hh=1) own points v+8*hh / 16+v+8*hh.
    float2* out2 = (float2*)out;
#pragma unroll
    for (int v = 0; v < 8; ++v) {
        if ((lane & 15) == v) {
            const int p0 = waveBase + v + 8 * hh;
            const int p1 = waveBase + 16 + v + 8 * hh;
            out2[p0] = make_float2(U0[v] / S0[v], V0[v] / S0[v]);
            out2[p1] = make_float2(U1[v] / S1[v], V1[v] / S1[v]);
        }
    }
}

extern "C" void kernel_launch(void* const* d_in, const int* in_sizes, int n_in,
                              void* d_out, int out_size, void* d_ws, size_t ws_size,
                              hipStream_t stream) {
    (void)n_in; (void)out_size; (void)d_ws; (void)ws_size;
    const float* xin        = (const float*)d_in[0];
    const float* centres    = (const float*)d_in[1];
    const float* log_sigmas = (const float*)d_in[2];
    const float* ax         = (const float*)d_in[3];
    const float* ay         = (const float*)d_in[4];
    float* out              = (float*)d_out;

    const int B = in_sizes[0] / 2;          // 65536
    const int blocks = (B + 255) / 256;     // 256 blocks, 256 points each
    RBF_kernel<<<blocks, 256, 0, stream>>>(xin, centres, log_sigmas, ax, ay, out);
}